// Op_43224550867568
// MI455X (gfx1250) — compile-verified
//
#include <hip/hip_runtime.h>
#include <stdint.h>

#define THREADS 256
#define WAVES_PER_BLOCK (THREADS / 32)
#define CHUNK 32

// ---------------------------------------------------------------------------
// Zero d_out (harness poisons it with 0xAA; we accumulate with atomics).
// ---------------------------------------------------------------------------
__global__ __launch_bounds__(THREADS) void zero_kernel(float* __restrict__ out, int n) {
  int n4 = n >> 2;
  float4* out4 = reinterpret_cast<float4*>(out);
  int i = blockIdx.x * blockDim.x + threadIdx.x;
  int stride = gridDim.x * blockDim.x;
  float4 z; z.x = 0.f; z.y = 0.f; z.z = 0.f; z.w = 0.f;
  for (; i < n4; i += stride) out4[i] = z;
  for (int t = (n4 << 2) + blockIdx.x * blockDim.x + threadIdx.x; t < n; t += stride)
    out[t] = 0.f;
}

// ---------------------------------------------------------------------------
// CDNA5 async global->LDS copy (gfx1250 VGLOBAL async op, ASYNCcnt-tracked).
// GVS mode: mem_addr = SADDR(64) + VADDR(32-bit byte offset).
// Metadata is single-use per launch -> non-temporal hint keeps L2 for x/out.
// ---------------------------------------------------------------------------
__device__ __forceinline__ void async_load_b32(unsigned lds_addr, unsigned byte_off,
                                               const void* sbase) {
  asm volatile("global_load_async_to_lds_b32 %0, %1, %2 th:TH_LOAD_NT"
               :: "v"(lds_addr), "v"(byte_off), "s"(sbase)
               : "memory");
}
__device__ __forceinline__ void wait_async_le0() {
  asm volatile("s_wait_asynccnt 0" ::: "memory");
}
__device__ __forceinline__ void wait_async_le3() {
  asm volatile("s_wait_asynccnt 3" ::: "memory");
}

__device__ __forceinline__ unsigned lds_off(const void* p) {
  // Low 32 bits of a flat shared-aperture pointer are the LDS byte offset.
  return (unsigned)(uintptr_t)p;
}

// ---------------------------------------------------------------------------
// Wave-per-edge SpMM-mixture: lane l handles dims [4l, 4l+4).
// Edge metadata double-buffered in LDS via async DMA; per-chunk x-row
// prefetch via global_prefetch_b8.
// ---------------------------------------------------------------------------
__global__ __launch_bounds__(THREADS)
void spmm_mix_kernel(const float* __restrict__ x,
                     const float* __restrict__ edge_vals,
                     const float* __restrict__ ws,
                     const int* __restrict__ edge_rows,
                     const int* __restrict__ edge_cols,
                     float* __restrict__ out,
                     int E, int totalEdges, float invNum) {
  __shared__ int   s_rows[WAVES_PER_BLOCK][2][CHUNK];
  __shared__ int   s_cols[WAVES_PER_BLOCK][2][CHUNK];
  __shared__ float s_vals[WAVES_PER_BLOCK][2][CHUNK];

  const int lane = threadIdx.x & 31;
  const int wid  = threadIdx.x >> 5;
  const int waveGlobal = blockIdx.x * WAVES_PER_BLOCK + wid;
  const int waveStride = gridDim.x * WAVES_PER_BLOCK;
  const int numChunks  = (totalEdges + CHUNK - 1) / CHUNK;

  auto issue = [&](int chunk, int buf) {
    unsigned g = (unsigned)chunk * CHUNK + (unsigned)lane;
    if (g < (unsigned)totalEdges) {
      unsigned off = g * 4u;
      async_load_b32(lds_off(&s_rows[wid][buf][lane]), off, edge_rows);
      async_load_b32(lds_off(&s_cols[wid][buf][lane]), off, edge_cols);
      async_load_b32(lds_off(&s_vals[wid][buf][lane]), off, edge_vals);
    }
  };

  int chunk = waveGlobal;
  int buf = 0;
  if (chunk < numChunks) issue(chunk, 0);

  while (chunk < numChunks) {
    const int next = chunk + waveStride;
    const bool hasNext = next < numChunks;
    if (hasNext) issue(next, buf ^ 1);

    // Overlap: wait only for THIS chunk's 3 async ops (next's 3 still fly).
    if (hasNext) wait_async_le3(); else wait_async_le0();

    const int base = chunk * CHUNK;
    int nEdges = totalEdges - base;
    if (nEdges > CHUNK) nEdges = CHUNK;

    // Prefetch all x rows of this chunk: lane l prefetches its own edge's
    // 512B row in 4 cacheline-granular global_prefetch_b8 ops.
    if (lane < nEdges) {
      const char* rowp =
          (const char*)(x + (size_t)s_cols[wid][buf][lane] * 128u);
      __builtin_prefetch(rowp, 0, 3);
      __builtin_prefetch(rowp + 128, 0, 3);
      __builtin_prefetch(rowp + 256, 0, 3);
      __builtin_prefetch(rowp + 384, 0, 3);
    }

    // Hoist op index / weight out of the edge loop when the chunk lies in a
    // single op (always true when E % CHUNK == 0); generic fallback otherwise.
    const unsigned opFirst = (unsigned)base / (unsigned)E;
    const unsigned opLast  = (unsigned)(base + nEdges - 1) / (unsigned)E;

    if (opFirst == opLast) {
      const float sw = ws[opFirst] * invNum;
      for (int e = 0; e < nEdges; ++e) {
        const int   row = s_rows[wid][buf][e];
        const int   col = s_cols[wid][buf][e];
        const float scale = s_vals[wid][buf][e] * sw;

        const float4 xv =
            *(reinterpret_cast<const float4*>(x + (size_t)col * 128u) + lane);
        float* orow = out + (size_t)row * 128u + (size_t)(lane << 2);
        unsafeAtomicAdd(orow + 0, scale * xv.x);
        unsafeAtomicAdd(orow + 1, scale * xv.y);
        unsafeAtomicAdd(orow + 2, scale * xv.z);
        unsafeAtomicAdd(orow + 3, scale * xv.w);
      }
    } else {
      for (int e = 0; e < nEdges; ++e) {
        const int   row = s_rows[wid][buf][e];
        const int   col = s_cols[wid][buf][e];
        const unsigned op = (unsigned)(base + e) / (unsigned)E;
        const float scale = s_vals[wid][buf][e] * ws[op] * invNum;

        const float4 xv =
            *(reinterpret_cast<const float4*>(x + (size_t)col * 128u) + lane);
        float* orow = out + (size_t)row * 128u + (size_t)(lane << 2);
        unsafeAtomicAdd(orow + 0, scale * xv.x);
        unsafeAtomicAdd(orow + 1, scale * xv.y);
        unsafeAtomicAdd(orow + 2, scale * xv.z);
        unsafeAtomicAdd(orow + 3, scale * xv.w);
      }
    }

    chunk = next;
    buf ^= 1;
  }
}

// ---------------------------------------------------------------------------
// Inputs (setup_inputs order): x[N*128] f32, edge_vals[8*E] f32, ws[8] f32,
// edge_rows[8*E] i32, edge_cols[8*E] i32. Output: [N*128] f32.
// ---------------------------------------------------------------------------
extern "C" void kernel_launch(void* const* d_in, const int* in_sizes, int n_in,
                              void* d_out, int out_size, void* d_ws, size_t ws_size,
                              hipStream_t stream) {
  const float* x         = (const float*)d_in[0];
  const float* edge_vals = (const float*)d_in[1];
  const float* wsv       = (const float*)d_in[2];
  const int*   edge_rows = (const int*)d_in[3];
  const int*   edge_cols = (const int*)d_in[4];
  float*       out       = (float*)d_out;

  const int num_op     = in_sizes[2];
  const int totalEdges = in_sizes[1];
  const int E          = totalEdges / num_op;
  // K = 1 -> num = num_op // 1 == num_op
  const float invNum = 1.0f / (float)num_op;

  zero_kernel<<<1024, THREADS, 0, stream>>>(out, out_size);

  const int blocks = 2048;  // persistent waves, ~6 chunks each
  spmm_mix_kernel<<<blocks, THREADS, 0, stream>>>(
      x, edge_vals, wsv, edge_rows, edge_cols, out, E, totalEdges, invNum);
}